// SplineconvModel_89361089560589
// MI455X (gfx1250) — compile-verified
//
#include <hip/hip_runtime.h>

typedef __attribute__((ext_vector_type(2))) float v2f;
typedef __attribute__((ext_vector_type(8))) float v8f;

#define N_NODES 100000
#define N_EDGES 1200000

// ws layout (floats):
//   [0, 12.8M)      y : [N_NODES][128]  = (x@W0 | x@W1)
//   [12.8M, 19.2M)  agg : [N_NODES][64]
//   [19.2M, 19.3M)  deg : [N_NODES]
#define WS_Y_OFF   0
#define WS_AGG_OFF (N_NODES * 128)
#define WS_DEG_OFF (N_NODES * 128 + N_NODES * 64)

// ---------------------------------------------------------------------------
// Kernel 1: fused triple GEMM  [100000 x 64] x [64 x 192] via fp32 WMMA.
// One wave32 per 16x16 output tile; K=64 -> 16 chained v_wmma_f32_16x16x4_f32.
// N-tiles 0..7 -> y (ws), N-tiles 8..11 -> d_out (x@root + bias).
// ---------------------------------------------------------------------------
__global__ __launch_bounds__(256) void spline_gemm_wmma(
    const float* __restrict__ x,     // [N,64]
    const float* __restrict__ W,     // [2,64,64] (W0 then W1)
    const float* __restrict__ root,  // [64,64]
    const float* __restrict__ bias,  // [64]
    float* __restrict__ y,           // [N,128] in ws
    float* __restrict__ out)         // [N,64]
{
    const int lane = threadIdx.x & 31;
    const int wave = threadIdx.x >> 5;
    const int tile = blockIdx.x * 8 + wave;   // 75000 tiles total, exact
    const int mtile = tile / 12;              // 0..6249
    const int ntile = tile % 12;              // 0..11

    const int half = lane >> 4;               // 0: lanes 0-15, 1: lanes 16-31
    const int l15  = lane & 15;
    const int kSel = half * 2;                // A/B upper-lane K offset

    // A fragment source: row = mtile*16 + l15, lanes>=16 hold K+2/K+3
    const float* A = x + (size_t)(mtile * 16 + l15) * 64;

    // B matrix select: W0 / W1 / root, all ld=64, col = lane's N
    const float* B;
    if (ntile < 4)      { B = W; }
    else if (ntile < 8) { B = W + 64 * 64; }
    else                { B = root; }
    const int col = (ntile & 3) * 16 + l15;

    v8f c = {};
#pragma unroll
    for (int kk = 0; kk < 16; ++kk) {
        const int k = kk * 4 + kSel;
        v2f a, b;
        a.x = A[k];                 // K = k   -> VGPR0
        a.y = A[k + 1];             // K = k+1 -> VGPR1
        b.x = B[k * 64 + col];      // B row K = k
        b.y = B[(k + 1) * 64 + col];
        // 8 args: (neg_a, A, neg_b, B, c_mod, C, reuse_a, reuse_b)
        c = __builtin_amdgcn_wmma_f32_16x16x4_f32(
                false, a, false, b, (short)0, c, false, false);
    }

    // C/D layout: VGPR r holds M = r (lanes 0-15) / M = r+8 (lanes 16-31)
    const int rowBase = mtile * 16 + half * 8;
    if (ntile < 8) {
        const int cc = ntile * 16 + l15;      // 0..127 into y
#pragma unroll
        for (int r = 0; r < 8; ++r)
            y[(size_t)(rowBase + r) * 128 + cc] = c[r];
    } else {
        const int cc = (ntile - 8) * 16 + l15; // 0..63 into out
        const float bv = bias[cc];
#pragma unroll
        for (int r = 0; r < 8; ++r)
            out[(size_t)(rowBase + r) * 64 + cc] = c[r] + bv;
    }
}

// ---------------------------------------------------------------------------
// Kernel 0: zero agg + deg in workspace
// ---------------------------------------------------------------------------
__global__ void zero_accum(float* __restrict__ p, int n) {
    int i = blockIdx.x * blockDim.x + threadIdx.x;
    if (i < n) p[i] = 0.0f;
}

// ---------------------------------------------------------------------------
// Kernel 2: per-edge blend + scatter-add. 64 threads per edge (one per
// channel). Gathers hit the 51MB y table resident in the 192MB L2.
// ---------------------------------------------------------------------------
__global__ __launch_bounds__(256) void edge_scatter(
    const int* __restrict__ ei,     // [2, E]
    const float* __restrict__ attr, // [E]
    const float* __restrict__ y,    // [N,128]
    float* __restrict__ agg,        // [N,64]
    float* __restrict__ deg)        // [N]
{
    const int t = blockIdx.x * 256 + threadIdx.x;
    const int e  = t >> 6;
    const int ch = t & 63;
    if (e >= N_EDGES) return;

    const int src = ei[e];
    const int dst = ei[N_EDGES + e];
    const float f = attr[e];        // K=2 -> frac == attr, lo == 0

    const float* ys = y + (size_t)src * 128;
    const float y0 = ys[ch];
    const float y1 = ys[64 + ch];
    const float m = (1.0f - f) * y0 + f * y1;

    atomicAdd(agg + (size_t)dst * 64 + ch, m);
    if (ch == 0) atomicAdd(deg + dst, 1.0f);
}

// ---------------------------------------------------------------------------
// Kernel 3: out = relu(agg / max(deg,1) + (x@root + bias already in out))
// ---------------------------------------------------------------------------
__global__ void finalize_relu(float* __restrict__ out,
                              const float* __restrict__ agg,
                              const float* __restrict__ deg) {
    int i = blockIdx.x * blockDim.x + threadIdx.x;
    if (i >= N_NODES * 64) return;
    float d = deg[i >> 6];
    d = d > 1.0f ? d : 1.0f;
    float v = out[i] + agg[i] / d;
    out[i] = v > 0.0f ? v : 0.0f;
}

extern "C" void kernel_launch(void* const* d_in, const int* in_sizes, int n_in,
                              void* d_out, int out_size, void* d_ws, size_t ws_size,
                              hipStream_t stream) {
    const float* x    = (const float*)d_in[0];
    const int*   ei   = (const int*)  d_in[1];
    const float* attr = (const float*)d_in[2];
    const float* W    = (const float*)d_in[3];
    const float* root = (const float*)d_in[4];
    const float* bias = (const float*)d_in[5];
    float* out = (float*)d_out;

    float* wsf = (float*)d_ws;
    float* y   = wsf + WS_Y_OFF;
    float* agg = wsf + WS_AGG_OFF;
    float* deg = wsf + WS_DEG_OFF;

    // 0) zero agg + deg
    {
        int n = N_NODES * 64 + N_NODES;
        zero_accum<<<(n + 255) / 256, 256, 0, stream>>>(agg, n);
    }
    // 1) WMMA GEMMs: 6250 M-tiles * 12 N-tiles = 75000 tiles, 8 waves/block
    spline_gemm_wmma<<<75000 / 8, 256, 0, stream>>>(x, W, root, bias, y, out);
    // 2) edge blend + scatter
    {
        long long threads = (long long)N_EDGES * 64;
        edge_scatter<<<(int)(threads / 256), 256, 0, stream>>>(ei, attr, y, agg, deg);
    }
    // 3) mean, residual add, relu
    finalize_relu<<<(N_NODES * 64) / 256, 256, 0, stream>>>(out, agg, deg);
}